// ISGNN_9586367004882
// MI455X (gfx1250) — compile-verified
//
#include <hip/hip_runtime.h>

#define NN 8192
#define EE 131072
#define GG 32

typedef float v2f __attribute__((ext_vector_type(2)));
typedef float v8f __attribute__((ext_vector_type(8)));

// ---------------- utility kernels ----------------

__global__ void k_zero(float* p, int n) {
  int i = blockIdx.x * blockDim.x + threadIdx.x;
  if (i < n) p[i] = 0.0f;
}

__global__ void k_count(const int* __restrict__ batch, float* counts, int n) {
  int i = blockIdx.x * blockDim.x + threadIdx.x;
  if (i < n) atomicAdd(&counts[batch[i]], 1.0f);
}

__global__ void k_scan(const float* __restrict__ counts, float* cum) {
  float acc = 0.f;
  for (int g = 0; g < GG; ++g) { cum[g] = acc; acc += counts[g]; }
  cum[GG] = acc;
}

__global__ void k_scatter1(const float* __restrict__ h, const int* __restrict__ src,
                           const int* __restrict__ dst, float* agg, int e) {
  int i = blockIdx.x * blockDim.x + threadIdx.x;
  if (i < e) atomicAdd(&agg[dst[i]], h[src[i]]);
}

// feature-dim = 2^logd scatter-add over edges
__global__ void k_scatterD(const float* __restrict__ h, const int* __restrict__ src,
                           const int* __restrict__ dst, float* agg, int e, int logd) {
  int i = blockIdx.x * blockDim.x + threadIdx.x;
  int tot = e << logd;
  if (i >= tot) return;
  int ed = i >> logd;
  int f  = i & ((1 << logd) - 1);
  int D  = 1 << logd;
  atomicAdd(&agg[dst[ed] * D + f], h[src[ed] * D + f]);
}

__global__ void k_add(const float* __restrict__ a, const float* __restrict__ b,
                      float* __restrict__ o, int n) {
  int i = blockIdx.x * blockDim.x + threadIdx.x;
  if (i < n) o[i] = a[i] + b[i];
}

// ---------------- InfoCollect 1->8->8->1 ----------------

__global__ void k_ic1(const float* __restrict__ inp, const float* __restrict__ agg,
                      const float* __restrict__ W, float* __restrict__ out, int n) {
  int i = blockIdx.x * blockDim.x + threadIdx.x;
  if (i >= n) return;
  float v = inp[i] + agg[i];
#pragma unroll
  for (int k = 0; k < 8; ++k) {
    float o = v * W[k];
    out[i * 8 + k] = o > 0.f ? o : 0.f;
  }
}

__global__ void k_ic2(const float* __restrict__ h, const float* __restrict__ agg,
                      const float* __restrict__ W, float* __restrict__ out, int n) {
  int i = blockIdx.x * blockDim.x + threadIdx.x;
  if (i >= n) return;
  float in[8];
#pragma unroll
  for (int j = 0; j < 8; ++j) in[j] = h[i * 8 + j] + agg[i * 8 + j];
#pragma unroll
  for (int k = 0; k < 8; ++k) {
    float acc = 0.f;
#pragma unroll
    for (int j = 0; j < 8; ++j) acc += in[j] * W[j * 8 + k];
    out[i * 8 + k] = acc > 0.f ? acc : 0.f;
  }
}

__global__ void k_ic3(const float* __restrict__ h, const float* __restrict__ agg,
                      const float* __restrict__ W, float* __restrict__ out, int n) {
  int i = blockIdx.x * blockDim.x + threadIdx.x;
  if (i >= n) return;
  float acc = 0.f;
#pragma unroll
  for (int j = 0; j < 8; ++j) acc += (h[i * 8 + j] + agg[i * 8 + j]) * W[j];
  out[i] = acc > 0.f ? acc : 0.f;
}

// ---------------- N x N tanh-sign reduction ----------------
// Dsum[i] += sum_{j in chunk} tanh(1000*(h[i]-h[j]) - 5).
// 2D grid: blockIdx.x tiles i (256 wide), blockIdx.y tiles j (1024-wide LDS chunk).
// 256 blocks keep the whole chip's TRANS pipes busy; Dsum must be zeroed first.
__global__ void k_pairwise(const float* __restrict__ h, float* __restrict__ Dsum, int n) {
  __shared__ float sh[1024];
  int jbase = blockIdx.y * 1024;
  for (int j = threadIdx.x; j < 1024; j += blockDim.x) sh[j] = h[jbase + j];
  __syncthreads();
  int i = blockIdx.x * blockDim.x + threadIdx.x;
  if (i >= n) return;
  float hi = h[i];
  float acc = 0.f;
#pragma unroll 4
  for (int j = 0; j < 1024; ++j) {
    float x = 1000.0f * (hi - sh[j]) - 5.0f;
    // tanh(x) = 1 - 2/(exp(2x)+1); saturates correctly at +/-1
    acc += 1.0f - 2.0f / (__expf(2.0f * x) + 1.0f);
  }
  atomicAdd(&Dsum[i], acc);
}

__global__ void k_f5(const float* __restrict__ Dsum, const int* __restrict__ batch,
                     const float* __restrict__ counts, const float* __restrict__ pa,
                     const float* __restrict__ pt, float* __restrict__ y, int n) {
  int i = blockIdx.x * blockDim.x + threadIdx.x;
  if (i >= n) return;
  float ng = counts[batch[i]];
  float v = pa[0] * (Dsum[i] / ng - pt[0]);
  y[i] = 1.f / (1.f + __expf(-v));
}

// ---------------- IndiLayer + segmented softmaxes ----------------

__global__ void k_indi(const float* __restrict__ y, const float* __restrict__ agg,
                       const float* __restrict__ W, float* __restrict__ r, int n) {
  int i = blockIdx.x * blockDim.x + threadIdx.x;
  if (i >= n) return;
  float o = (y[i] + agg[i]) * W[0];
  r[i] = o > 0.f ? o : 0.f;
}

// logits are >= 0 after relu, so int-bit atomicMax is order-preserving; m zero-inited.
__global__ void k_segmax(const float* __restrict__ r, const int* __restrict__ batch,
                         int* m1, int* m2, int n) {
  int i = blockIdx.x * blockDim.x + threadIdx.x;
  if (i >= n) return;
  int g = batch[i];
  float rr = r[i];
  atomicMax(&m1[g], __float_as_int(rr * 10.f));
  atomicMax(&m2[g], __float_as_int(rr * 2.f));
}

__global__ void k_segexp(const float* __restrict__ r, const int* __restrict__ batch,
                         const float* __restrict__ m1, const float* __restrict__ m2,
                         float* s1, float* s2, int n) {
  int i = blockIdx.x * blockDim.x + threadIdx.x;
  if (i >= n) return;
  int g = batch[i];
  float rr = r[i];
  atomicAdd(&s1[g], __expf(rr * 10.f - m1[g]));
  atomicAdd(&s2[g], __expf(rr * 2.f - m2[g]));
}

__global__ void k_indifin(const float* __restrict__ r, const int* __restrict__ batch,
                          const float* __restrict__ m1, const float* __restrict__ s1,
                          const float* __restrict__ m2, const float* __restrict__ s2,
                          const float* __restrict__ cum, float* __restrict__ y,
                          float* outidx, int n) {
  int i = blockIdx.x * blockDim.x + threadIdx.x;
  if (i >= n) return;
  int g = batch[i];
  float rr = r[i];
  float p1 = __expf(rr * 10.f - m1[g]) / s1[g];
  float pgi = (float)i - cum[g];
  atomicAdd(&outidx[g], p1 * pgi);
  float p2 = __expf(rr * 2.f - m2[g]) / s2[g];
  y[i] = p2 * rr + y[i];
}

// ---------------- GIN stack ----------------

// out[n,k] = relu((y[n]+agg[n]) * Wa0[k]), Wa0 is (1,64)
__global__ void k_expand(const float* __restrict__ y, const float* __restrict__ agg,
                         const float* __restrict__ W, float* __restrict__ out, int n) {
  int i = blockIdx.x * blockDim.x + threadIdx.x;
  if (i >= n * 64) return;
  int node = i >> 6, k = i & 63;
  float o = (y[node] + agg[node]) * W[k];
  out[i] = o > 0.f ? o : 0.f;
}

// C = relu(A @ W); A:[nrows,64] fp32, W:[64,64] fp32, exact fp32 via
// V_WMMA_F32_16X16X4_F32. Block = 8 waves, 128 rows. W (16KB) is staged once
// per block in LDS; each wave computes its full 16x64 output band (4 column
// tiles) so A fragments are reused 4x from registers and W comes from ds_load.
// 64 chained WMMAs per wave.
__global__ void __launch_bounds__(256) k_wmma_relu(const float* __restrict__ A,
                                                   const float* __restrict__ W,
                                                   float* __restrict__ C, int nrows) {
  __shared__ float shW[64 * 64];
  for (int t = threadIdx.x; t < 64 * 64; t += 256) shW[t] = W[t];
  __syncthreads();

  int wave = threadIdx.x >> 5;
  int lane = threadIdx.x & 31;
  int m0 = blockIdx.x * 128 + wave * 16;
  if (m0 >= nrows) return;     // wave-uniform (never splits EXEC)
  int half = lane >> 4;        // 0: K,K+1   1: K+2,K+3
  int l = lane & 15;

  v8f c0 = {0.f,0.f,0.f,0.f,0.f,0.f,0.f,0.f};
  v8f c1 = c0, c2 = c0, c3 = c0;
#pragma unroll
  for (int k = 0; k < 64; k += 4) {
    int ka = k + 2 * half;
    v2f a;
    a.x = A[(m0 + l) * 64 + ka];
    a.y = A[(m0 + l) * 64 + ka + 1];
    v2f b0, b1, b2, b3;
    b0.x = shW[ka * 64 + l];        b0.y = shW[(ka + 1) * 64 + l];
    b1.x = shW[ka * 64 + 16 + l];   b1.y = shW[(ka + 1) * 64 + 16 + l];
    b2.x = shW[ka * 64 + 32 + l];   b2.y = shW[(ka + 1) * 64 + 32 + l];
    b3.x = shW[ka * 64 + 48 + l];   b3.y = shW[(ka + 1) * 64 + 48 + l];
    c0 = __builtin_amdgcn_wmma_f32_16x16x4_f32(false, a, false, b0, (short)0, c0, false, false);
    c1 = __builtin_amdgcn_wmma_f32_16x16x4_f32(false, a, false, b1, (short)0, c1, false, false);
    c2 = __builtin_amdgcn_wmma_f32_16x16x4_f32(false, a, false, b2, (short)0, c2, false, false);
    c3 = __builtin_amdgcn_wmma_f32_16x16x4_f32(false, a, false, b3, (short)0, c3, false, false);
  }
#pragma unroll
  for (int v = 0; v < 8; ++v) {
    int row = (m0 + v + 8 * half) * 64;
    float v0 = c0[v], v1 = c1[v], v2 = c2[v], v3 = c3[v];
    C[row + l]      = v0 > 0.f ? v0 : 0.f;
    C[row + 16 + l] = v1 > 0.f ? v1 : 0.f;
    C[row + 32 + l] = v2 > 0.f ? v2 : 0.f;
    C[row + 48 + l] = v3 > 0.f ? v3 : 0.f;
  }
}

// res[g,:] += h[n,:] @ Wout (64x16), one thread per (node, out-col)
__global__ void k_pool(const float* __restrict__ h, const float* __restrict__ W,
                       const int* __restrict__ batch, float* out, int n) {
  int i = blockIdx.x * blockDim.x + threadIdx.x;
  if (i >= n * 16) return;
  int node = i >> 4, o = i & 15;
  float acc = 0.f;
#pragma unroll
  for (int k = 0; k < 64; ++k) acc += h[node * 64 + k] * W[k * 16 + o];
  atomicAdd(&out[batch[node] * 16 + o], acc);
}

// ---------------- host orchestration ----------------

extern "C" void kernel_launch(void* const* d_in, const int* in_sizes, int n_in,
                              void* d_out, int out_size, void* d_ws, size_t ws_size,
                              hipStream_t stream) {
  const float* x     = (const float*)d_in[0];
  const int*   eidx  = (const int*)d_in[1];
  const int*   src   = eidx;
  const int*   dst   = eidx + EE;
  const int*   batch = (const int*)d_in[2];
  const float* Wic1  = (const float*)d_in[3];
  const float* Wic2  = (const float*)d_in[4];
  const float* Wic3  = (const float*)d_in[5];
  const float* Windi = (const float*)d_in[6];
  const float* f5a   = (const float*)d_in[7];
  const float* f5t   = (const float*)d_in[8];
  const float* Wa0   = (const float*)d_in[9];
  const float* Wb0   = (const float*)d_in[10];
  const float* Wa1   = (const float*)d_in[11];
  const float* Wb1   = (const float*)d_in[12];
  const float* Wa2   = (const float*)d_in[13];
  const float* Wb2   = (const float*)d_in[14];
  const float* Wout  = (const float*)d_in[15];
  float* out = (float*)d_out;

  float* ws     = (float*)d_ws;
  float* y      = ws;              // N
  float* h1     = ws + 8192;       // N*8
  float* h2     = ws + 73728;      // N*8
  float* h3     = ws + 139264;     // N
  float* agg1   = ws + 147456;     // N
  float* agg8   = ws + 155648;     // N*8
  float* Dsum   = ws + 221184;     // N
  float* reluv  = ws + 229376;     // N
  float* counts = ws + 237568;     // 32
  float* cum    = ws + 237600;     // 33
  float* m1     = ws + 237664;     // 32 (m1,s1,m2,s2 contiguous 128)
  float* s1     = ws + 237696;
  float* m2     = ws + 237728;
  float* s2     = ws + 237760;
  float* fA     = ws + 237792;     // N*64
  float* fB     = ws + 762080;     // N*64
  float* agg64  = ws + 1286368;    // N*64

  auto nb = [](int n) { return (n + 255) / 256; };
  dim3 pairgrid(NN / 256, NN / 1024);   // 32 x 8 = 256 blocks
  int wmma_blocks = NN / 128;           // 8 waves x 16 rows per block

  // init (re-done every call: deterministic under graph replay)
  k_zero<<<nb(out_size), 256, 0, stream>>>(out, out_size);
  k_zero<<<1, 256, 0, stream>>>(counts, 32);
  k_count<<<nb(NN), 256, 0, stream>>>(batch, counts, NN);
  k_scan<<<1, 1, 0, stream>>>(counts, cum);

  auto ic_block = [&](const float* inp) {
    k_zero<<<nb(NN), 256, 0, stream>>>(agg1, NN);
    k_scatter1<<<nb(EE), 256, 0, stream>>>(inp, src, dst, agg1, EE);
    k_ic1<<<nb(NN), 256, 0, stream>>>(inp, agg1, Wic1, h1, NN);
    k_zero<<<nb(NN * 8), 256, 0, stream>>>(agg8, NN * 8);
    k_scatterD<<<nb(EE * 8), 256, 0, stream>>>(h1, src, dst, agg8, EE, 3);
    k_ic2<<<nb(NN), 256, 0, stream>>>(h1, agg8, Wic2, h2, NN);
    k_zero<<<nb(NN * 8), 256, 0, stream>>>(agg8, NN * 8);
    k_scatterD<<<nb(EE * 8), 256, 0, stream>>>(h2, src, dst, agg8, EE, 3);
    k_ic3<<<nb(NN), 256, 0, stream>>>(h2, agg8, Wic3, h3, NN);
    k_zero<<<nb(NN), 256, 0, stream>>>(Dsum, NN);
    k_pairwise<<<pairgrid, 256, 0, stream>>>(h3, Dsum, NN);
    k_f5<<<nb(NN), 256, 0, stream>>>(Dsum, batch, counts, f5a, f5t, y, NN);
  };

  for (int L = 0; L < 3; ++L) {
    ic_block(L == 0 ? x : y);
    k_zero<<<nb(NN), 256, 0, stream>>>(agg1, NN);
    k_scatter1<<<nb(EE), 256, 0, stream>>>(y, src, dst, agg1, EE);
    k_indi<<<nb(NN), 256, 0, stream>>>(y, agg1, Windi, reluv, NN);
    k_zero<<<1, 256, 0, stream>>>(m1, 128);
    k_segmax<<<nb(NN), 256, 0, stream>>>(reluv, batch, (int*)m1, (int*)m2, NN);
    k_segexp<<<nb(NN), 256, 0, stream>>>(reluv, batch, m1, m2, s1, s2, NN);
    k_indifin<<<nb(NN), 256, 0, stream>>>(reluv, batch, m1, s1, m2, s2, cum, y,
                                          out + GG * 16 + GG * L, NN);
  }
  ic_block(y);

  // GIN layer 0: h [N,1] -> [N,64]
  k_zero<<<nb(NN), 256, 0, stream>>>(agg1, NN);
  k_scatter1<<<nb(EE), 256, 0, stream>>>(y, src, dst, agg1, EE);
  k_expand<<<nb(NN * 64), 256, 0, stream>>>(y, agg1, Wa0, fA, NN);
  k_wmma_relu<<<wmma_blocks, 256, 0, stream>>>(fA, Wb0, fB, NN);  // h = fB

  // GIN layer 1
  k_zero<<<nb(NN * 64), 256, 0, stream>>>(agg64, NN * 64);
  k_scatterD<<<nb(EE * 64), 256, 0, stream>>>(fB, src, dst, agg64, EE, 6);
  k_add<<<nb(NN * 64), 256, 0, stream>>>(fB, agg64, fA, NN * 64);
  k_wmma_relu<<<wmma_blocks, 256, 0, stream>>>(fA, Wa1, fB, NN);
  k_wmma_relu<<<wmma_blocks, 256, 0, stream>>>(fB, Wb1, fA, NN);  // h = fA

  // GIN layer 2
  k_zero<<<nb(NN * 64), 256, 0, stream>>>(agg64, NN * 64);
  k_scatterD<<<nb(EE * 64), 256, 0, stream>>>(fA, src, dst, agg64, EE, 6);
  k_add<<<nb(NN * 64), 256, 0, stream>>>(fA, agg64, fB, NN * 64);
  k_wmma_relu<<<wmma_blocks, 256, 0, stream>>>(fB, Wa2, fA, NN);
  k_wmma_relu<<<wmma_blocks, 256, 0, stream>>>(fA, Wb2, fB, NN);  // h = fB

  // global_add_pool(h @ Wout)
  k_pool<<<nb(NN * 16), 256, 0, stream>>>(fB, Wout, batch, out, NN);
}